// SAGEConvPassMessage_xn_only_76192719831672
// MI455X (gfx1250) — compile-verified
//
#include <hip/hip_runtime.h>

typedef __attribute__((ext_vector_type(2))) float v2f;
typedef __attribute__((ext_vector_type(8))) float v8f;

#define DIM 64

// ---------------------------------------------------------------------------
// Kernel 1: zero the aggregation workspace (d_ws is poisoned by the harness).
// ---------------------------------------------------------------------------
__global__ void zero_ws(float* __restrict__ p, long long n) {
    long long i4 = ((long long)blockIdx.x * blockDim.x + threadIdx.x) * 4;
    if (i4 + 3 < n) {
        *(float4*)(p + i4) = make_float4(0.f, 0.f, 0.f, 0.f);
    } else {
        for (long long k = i4; k < n; ++k) p[k] = 0.f;
    }
}

// ---------------------------------------------------------------------------
// Kernel 2: edge scatter-add (segment_sum). 16 threads per edge, float4 loads
// of the source row (coalesced), 4 global f32 atomics into the dest row.
// The 25.6MB accumulator stays L2-resident (192MB L2), so atomics are cheap.
// ---------------------------------------------------------------------------
__global__ void scatter_edges(const float* __restrict__ hn,
                              const int*   __restrict__ src,
                              const int*   __restrict__ dst,
                              float*       __restrict__ aggr,
                              int n_edges) {
    long long tid = (long long)blockIdx.x * blockDim.x + threadIdx.x;
    int e = (int)(tid >> 4);
    int q = (int)(tid & 15);
    if (e >= n_edges) return;
    int s = src[e];
    int d = dst[e];
    const float4 v = *((const float4*)(hn + (long long)s * DIM) + q);
    float* o = aggr + (long long)d * DIM + q * 4;
    atomicAdd(o + 0, v.x);
    atomicAdd(o + 1, v.y);
    atomicAdd(o + 2, v.z);
    atomicAdd(o + 3, v.w);
}

// ---------------------------------------------------------------------------
// Kernel 3: fused [hn | aggr] @ W + b via V_WMMA_F32_16X16X4_F32.
// 256 threads = 8 wave32s; each wave computes a 16x64 output tile.
// W is staged transposed in LDS once per block; A fragments come straight
// from global (16 rows x 512B working set per wave -> cache resident).
//
// ISA 7.12.2 32-bit layouts (wave32):
//   A 16x4 : lane L holds M=L%16; VGPR0 = K=(L<16?0:2), VGPR1 = K+1
//   B 4x16 : lane L holds N=L%16; VGPR0 = K=(L<16?0:2), VGPR1 = K+1
//   C 16x16: VGPR j, lane L -> M = j + (L<16?0:8), N = L%16
// ---------------------------------------------------------------------------
__global__ __launch_bounds__(256)
void sage_gemm_wmma(const float* __restrict__ hn,
                    const float* __restrict__ aggr,
                    const float* __restrict__ W,
                    const float* __restrict__ bias,
                    float*       __restrict__ out,
                    int n_nodes) {
    __shared__ float sWt[64][130];   // W transposed: sWt[n][k], padded stride
    __shared__ float sB[64];

    int t = threadIdx.x;

    // Stage W (128x64, row-major) transposed into LDS. Coalesced global reads.
#pragma unroll
    for (int i = 0; i < 32; ++i) {
        int idx = i * 256 + t;          // 0..8191
        int k = idx >> 6;
        int n = idx & 63;
        sWt[n][k] = W[idx];
    }
    if (t < 64) sB[t] = bias[t];
    __syncthreads();

    int wave = t >> 5;
    int lane = t & 31;
    int l16  = lane & 15;
    int hi   = lane >> 4;               // half-wave: selects K-pair / M+8

    long long rowbase = (long long)blockIdx.x * 128 + wave * 16;
    long long row  = rowbase + l16;
    long long rowc = (row < n_nodes) ? row : (long long)(n_nodes - 1); // safe clamp

    v8f acc0 = {}, acc1 = {}, acc2 = {}, acc3 = {};

    // Two K=64 passes: columns 0..63 from hn, 64..127 from aggr (fused concat).
#pragma unroll
    for (int hm = 0; hm < 2; ++hm) {
        const float* rp = (hm ? aggr : hn) + rowc * DIM;
#pragma unroll
        for (int kc = 0; kc < 64; kc += 4) {
            int kg = kc + 2 * hi;                    // this lane's K pair
            v2f a = *(const v2f*)(rp + kg);          // 8B-aligned global load
            int kW = hm * 64 + kg;
            v2f b0 = *(const v2f*)(&sWt[ 0 + l16][kW]);  // 8B ds loads
            v2f b1 = *(const v2f*)(&sWt[16 + l16][kW]);
            v2f b2 = *(const v2f*)(&sWt[32 + l16][kW]);
            v2f b3 = *(const v2f*)(&sWt[48 + l16][kW]);
            acc0 = __builtin_amdgcn_wmma_f32_16x16x4_f32(false, a, false, b0, (short)0, acc0, false, false);
            acc1 = __builtin_amdgcn_wmma_f32_16x16x4_f32(false, a, false, b1, (short)0, acc1, false, false);
            acc2 = __builtin_amdgcn_wmma_f32_16x16x4_f32(false, a, false, b2, (short)0, acc2, false, false);
            acc3 = __builtin_amdgcn_wmma_f32_16x16x4_f32(false, a, false, b3, (short)0, acc3, false, false);
        }
    }

    // Bias + store. Each store instruction writes two contiguous 64B segments
    // (lanes 0-15 row m, lanes 16-31 row m+8).
    float bn0 = sB[ 0 + l16];
    float bn1 = sB[16 + l16];
    float bn2 = sB[32 + l16];
    float bn3 = sB[48 + l16];
#pragma unroll
    for (int j = 0; j < 8; ++j) {
        long long m = rowbase + j + hi * 8;
        if (m < n_nodes) {
            float* orow = out + m * DIM;
            orow[ 0 + l16] = acc0[j] + bn0;
            orow[16 + l16] = acc1[j] + bn1;
            orow[32 + l16] = acc2[j] + bn2;
            orow[48 + l16] = acc3[j] + bn3;
        }
    }
}

// ---------------------------------------------------------------------------
// Launcher
// Inputs (setup_inputs order): 0=hn [N*64] f32, 1=he (unused), 2=src [E] int,
//                              3=dst [E] int, 4=W [128*64] f32, 5=b [64] f32
// ---------------------------------------------------------------------------
extern "C" void kernel_launch(void* const* d_in, const int* in_sizes, int n_in,
                              void* d_out, int out_size, void* d_ws, size_t ws_size,
                              hipStream_t stream) {
    const float* hn  = (const float*)d_in[0];
    const int*   src = (const int*)  d_in[2];
    const int*   dst = (const int*)  d_in[3];
    const float* W   = (const float*)d_in[4];
    const float* b   = (const float*)d_in[5];
    float* out  = (float*)d_out;
    float* aggr = (float*)d_ws;      // [n_nodes * 64] f32 accumulator

    int n_nodes = in_sizes[0] / DIM;
    int n_edges = in_sizes[2];

    // 1) zero the accumulator
    long long zn = (long long)n_nodes * DIM;
    int zblocks = (int)((zn / 4 + 255) / 256);
    zero_ws<<<zblocks, 256, 0, stream>>>(aggr, zn);

    // 2) scatter-add messages
    long long sthreads = (long long)n_edges * 16;
    int sblocks = (int)((sthreads + 255) / 256);
    scatter_edges<<<sblocks, 256, 0, stream>>>(hn, src, dst, aggr, n_edges);

    // 3) fused concat + GEMM + bias via WMMA
    int gblocks = (n_nodes + 127) / 128;
    sage_gemm_wmma<<<gblocks, 256, 0, stream>>>(hn, aggr, W, b, out, n_nodes);
}